// KMLoss_57964878626830
// MI455X (gfx1250) — compile-verified
//
#include <hip/hip_runtime.h>
#include <hip/hip_bf16.h>

typedef __attribute__((ext_vector_type(2))) float v2f;
typedef __attribute__((ext_vector_type(8))) float v8f;

#define TOPK    256
#define NPTS    4096
#define NJ      256
#define NB      128
#define ITERS   25
#define GSTRIDE 257   // +1 col pad -> conflict-free column sweeps (257 mod 64 coprime)

__device__ __forceinline__ float wave_max_f(float v) {
#pragma unroll
    for (int m = 16; m >= 1; m >>= 1) v = fmaxf(v, __shfl_xor(v, m, 32));
    return v;
}
__device__ __forceinline__ float wave_sum_f(float v) {
#pragma unroll
    for (int m = 16; m >= 1; m >>= 1) v += __shfl_xor(v, m, 32);
    return v;
}
__device__ __forceinline__ int wave_sum_i(int v) {
#pragma unroll
    for (int m = 16; m >= 1; m >>= 1) v += __shfl_xor(v, m, 32);
    return v;
}

extern "C" __global__ __launch_bounds__(256)
void kmloss_batch(const float* __restrict__ pts,
                  const float* __restrict__ log_gamma,
                  const float* __restrict__ prob,
                  float* __restrict__ batch_loss)
{
    __shared__ int   s_idx[TOPK];
    __shared__ float s_xp[TOPK][4];   // gathered points, padded to K=4 for WMMA
    __shared__ float s_xx[TOPK];
    __shared__ float s_yp[NJ][4];     // mu, padded
    __shared__ float s_yy[NJ];
    __shared__ float s_logc[NJ];
    __shared__ float s_logu[TOPK];
    __shared__ float s_logv[NJ];
    __shared__ float s_rowlse[TOPK];  // max + log(sum exp) of log_score rows
    __shared__ int   s_scan[256];
    __shared__ int   s_redI[8];
    __shared__ float s_redF[8];
    extern __shared__ float s_dyn[];  // 256*257 floats (~257KB; CDNA5 WGP has 320KB LDS)

    const int t = threadIdx.x;
    const int w = t >> 5;
    const int l = t & 31;
    const int b = blockIdx.x;

    // ---------- Phase 0: top-256 of prob row via binary search on float bits ----------
    unsigned int ub[16];
    {
        const float* pb = prob + (size_t)b * NPTS;
#pragma unroll
        for (int e = 0; e < 16; e++) ub[e] = __float_as_uint(pb[t + 256 * e]);
    }
    unsigned int lo = 0u, hi = 0x3F800000u;  // prob in [0,1): bits monotone as uint
    while (lo < hi) {
        unsigned int mid = lo + ((hi - lo + 1u) >> 1);
        int c = 0;
#pragma unroll
        for (int e = 0; e < 16; e++) c += (ub[e] >= mid) ? 1 : 0;
        c = wave_sum_i(c);
        if (l == 0) s_redI[w] = c;
        __syncthreads();
        int tot = 0;
#pragma unroll
        for (int i = 0; i < 8; i++) tot += s_redI[i];
        __syncthreads();
        if (tot >= TOPK) lo = mid; else hi = mid - 1u;
    }
    const unsigned int tstar = lo;  // max threshold with count >= 256 (== 256 w/o ties)

    int cnt = 0;
#pragma unroll
    for (int e = 0; e < 16; e++) cnt += (ub[e] >= tstar) ? 1 : 0;
    s_scan[t] = cnt;
    __syncthreads();
    for (int off = 1; off < 256; off <<= 1) {   // Hillis-Steele inclusive scan
        int v = (t >= off) ? s_scan[t - off] : 0;
        __syncthreads();
        s_scan[t] += v;
        __syncthreads();
    }
    {
        int pos = s_scan[t] - cnt;              // exclusive prefix (deterministic order)
#pragma unroll
        for (int e = 0; e < 16; e++) {
            if (ub[e] >= tstar) {
                if (pos < TOPK) s_idx[pos] = t + 256 * e;
                pos++;
            }
        }
    }
    __syncthreads();

    // ---------- Phase 1: gather pts, row-softmax stats, pi / mu ----------
    {
        int id = s_idx[t];
        const float* pp = pts + ((size_t)b * NPTS + id) * 3;
        float p0 = pp[0], p1 = pp[1], p2 = pp[2];
        s_xp[t][0] = p0; s_xp[t][1] = p1; s_xp[t][2] = p2; s_xp[t][3] = 0.f;
        s_xx[t] = p0 * p0 + p1 * p1 + p2 * p2;
    }
    __syncthreads();

    float piacc[8], muacc[8][3];
#pragma unroll
    for (int c = 0; c < 8; c++) { piacc[c] = 0.f; muacc[c][0] = muacc[c][1] = muacc[c][2] = 0.f; }

    const size_t gbase = (size_t)b * NPTS * (size_t)NJ;
    for (int r = 0; r < 32; r++) {               // wave w owns rows [w*32, w*32+32)
        int k = w * 32 + r;
        const float* lsrow = log_gamma + gbase + (size_t)s_idx[k] * NJ;
        float ls[8];
#pragma unroll
        for (int c = 0; c < 8; c++) ls[c] = lsrow[l + 32 * c];
        float m = ls[0];
#pragma unroll
        for (int c = 1; c < 8; c++) m = fmaxf(m, ls[c]);
        m = wave_max_f(m);
        float s = 0.f;
#pragma unroll
        for (int c = 0; c < 8; c++) s += __expf(ls[c] - m);
        s = wave_sum_f(s);
        float lse = m + __logf(s);
        if (l == 0) s_rowlse[k] = lse;
        float x0 = s_xp[k][0], x1 = s_xp[k][1], x2 = s_xp[k][2];
#pragma unroll
        for (int c = 0; c < 8; c++) {
            float sc = __expf(ls[c] - lse);      // softmax(log_score) row k
            piacc[c] += sc;
            muacc[c][0] += sc * x0; muacc[c][1] += sc * x1; muacc[c][2] += sc * x2;
        }
    }
    // deterministic cross-wave reduction through the (not-yet-used) dynamic buffer
#pragma unroll
    for (int c = 0; c < 8; c++) {
        int j = l + 32 * c;
        float* dst = s_dyn + ((w * 256 + j) << 2);
        dst[0] = piacc[c]; dst[1] = muacc[c][0]; dst[2] = muacc[c][1]; dst[3] = muacc[c][2];
    }
    __syncthreads();
    {
        int j = t;
        float ps = 0.f, m0 = 0.f, m1 = 0.f, m2 = 0.f;
        for (int ww = 0; ww < 8; ww++) {
            const float* src = s_dyn + ((ww * 256 + j) << 2);
            ps += src[0]; m0 += src[1]; m1 += src[2]; m2 += src[3];
        }
        float pi  = fmaxf(ps * (1.f / 256.f), 1e-4f);   // clip(mean, 1e-4)
        float npi = pi * 256.f;
        m0 /= npi; m1 /= npi; m2 /= npi;
        s_yp[j][0] = m0; s_yp[j][1] = m1; s_yp[j][2] = m2; s_yp[j][3] = 0.f;
        s_yy[j] = m0 * m0 + m1 * m1 + m2 * m2;
        s_logc[j] = pi;                                  // temp: clipped pi
        s_logv[j] = 0.f;                                 // Sinkhorn init
    }
    __syncthreads();
    {
        float pij = s_logc[t];
        float v = wave_sum_f(pij);
        if (l == 0) s_redF[w] = v;
        __syncthreads();
        float tot = 0.f;
#pragma unroll
        for (int i = 0; i < 8; i++) tot += s_redF[i];
        s_logc[t] = __logf(fmaxf(pij / tot, 1e-9f));     // log column marginal
    }
    __syncthreads();

    // ---------- Phase 2: WMMA Gram matrix -> log_K in LDS ----------
    // log_K[k,j] = (2*x_k.y_j - xx[k] - yy[j]) / eps, eps = 0.1
    {
        const int half = l >> 4;
        const int ml   = l & 15;
        const int k0   = half * 2;                       // f32 16x16x4 A/B lane layout
        for (int tile = w; tile < 256; tile += 8) {      // 32 tiles per wave, EXEC all-1s
            int km = (tile >> 4) * 16, jn = (tile & 15) * 16;
            v2f a, bb;
            a.x  = s_xp[km + ml][k0];  a.y  = s_xp[km + ml][k0 + 1];
            bb.x = s_yp[jn + ml][k0];  bb.y = s_yp[jn + ml][k0 + 1];
            v8f cz = {0.f, 0.f, 0.f, 0.f, 0.f, 0.f, 0.f, 0.f};
            v8f d = __builtin_amdgcn_wmma_f32_16x16x4_f32(
                        false, a, false, bb, (short)0, cz, false, false);
            int j = jn + ml;
            float yyj = s_yy[j];
#pragma unroll
            for (int r = 0; r < 8; r++) {
                int k = km + r + half * 8;               // C/D layout: VGPR r, halves M / M+8
                s_dyn[k * GSTRIDE + j] = (2.f * d[r] - s_xx[k] - yyj) * 10.0f;
            }
        }
    }
    __syncthreads();

    // ---------- Phase 3: 25 Sinkhorn iterations ----------
    const float log_r = -5.545177444479562f;             // -log(256)
    for (int it = 0; it < ITERS; it++) {
        float lv[8];
#pragma unroll
        for (int c = 0; c < 8; c++) lv[c] = s_logv[l + 32 * c];
        for (int r = 0; r < 32; r++) {                   // row pass -> log_u
            int k = w * 32 + r;
            const float* gk = s_dyn + k * GSTRIDE;
            float tb[8];
#pragma unroll
            for (int c = 0; c < 8; c++) tb[c] = gk[l + 32 * c] + lv[c];
            float m = tb[0];
#pragma unroll
            for (int c = 1; c < 8; c++) m = fmaxf(m, tb[c]);
            m = wave_max_f(m);
            float s = 0.f;
#pragma unroll
            for (int c = 0; c < 8; c++) s += __expf(tb[c] - m);
            s = wave_sum_f(s);
            if (l == 0) s_logu[k] = log_r - (m + __logf(s));
        }
        __syncthreads();
        float lu[8];
#pragma unroll
        for (int c = 0; c < 8; c++) lu[c] = s_logu[l + 32 * c];
        for (int r = 0; r < 32; r++) {                   // col pass -> log_v (pad => no conflicts)
            int j = w * 32 + r;
            float tb[8];
#pragma unroll
            for (int c = 0; c < 8; c++) tb[c] = s_dyn[(l + 32 * c) * GSTRIDE + j] + lu[c];
            float m = tb[0];
#pragma unroll
            for (int c = 1; c < 8; c++) m = fmaxf(m, tb[c]);
            m = wave_max_f(m);
            float s = 0.f;
#pragma unroll
            for (int c = 0; c < 8; c++) s += __expf(tb[c] - m);
            s = wave_sum_f(s);
            if (l == 0) s_logv[j] = s_logc[j] - (m + __logf(s));
        }
        __syncthreads();
    }

    // ---------- Phase 4: loss = sum_kj -(T/rowsumT) * (log_score - rowLSE) ----------
    float acc = 0.f;
    {
        float lv[8];
#pragma unroll
        for (int c = 0; c < 8; c++) lv[c] = s_logv[l + 32 * c];
        for (int r = 0; r < 32; r++) {
            int k = w * 32 + r;
            float lu = s_logu[k];
            float rl = s_rowlse[k];
            const float* gk = s_dyn + k * GSTRIDE;
            const float* lsrow = log_gamma + gbase + (size_t)s_idx[k] * NJ;
            float ls[8], tv[8];
#pragma unroll
            for (int c = 0; c < 8; c++) {
                ls[c] = lsrow[l + 32 * c];
                tv[c] = __expf(lu + gk[l + 32 * c] + lv[c]);   // transport plan T
            }
            float s = 0.f;
#pragma unroll
            for (int c = 0; c < 8; c++) s += tv[c];
            s = wave_sum_f(s);
            float inv = 1.f / fmaxf(s, 1e-4f);
#pragma unroll
            for (int c = 0; c < 8; c++) acc += tv[c] * inv * (rl - ls[c]);
        }
    }
    acc = wave_sum_f(acc);
    if (l == 0) s_redF[w] = acc;
    __syncthreads();
    if (t == 0) {
        float tot = 0.f;
        for (int i = 0; i < 8; i++) tot += s_redF[i];
        batch_loss[b] = tot;
    }
}

extern "C" __global__ void kmloss_finalize(const float* __restrict__ batch_loss,
                                           float* __restrict__ out)
{
    if (threadIdx.x == 0 && blockIdx.x == 0) {
        float s = 0.f;
        for (int b = 0; b < NB; b++) s += batch_loss[b];   // fixed order: deterministic
        out[0] = s * (1.0f / (128.0f * 256.0f));           // mean over (B, J)
    }
}

extern "C" void kernel_launch(void* const* d_in, const int* in_sizes, int n_in,
                              void* d_out, int out_size, void* d_ws, size_t ws_size,
                              hipStream_t stream)
{
    (void)in_sizes; (void)n_in; (void)out_size; (void)ws_size;
    const float* pts       = (const float*)d_in[0];
    const float* log_gamma = (const float*)d_in[1];
    const float* prob      = (const float*)d_in[2];
    float* ws = (float*)d_ws;                              // 128 floats of scratch

    size_t shmem = (size_t)256 * GSTRIDE * sizeof(float);  // 263,168 B < 320 KB WGP LDS
    kmloss_batch<<<NB, 256, shmem, stream>>>(pts, log_gamma, prob, ws);
    kmloss_finalize<<<1, 32, 0, stream>>>(ws, (float*)d_out);
}